// GCNN_69776038691375
// MI455X (gfx1250) — compile-verified
//
#include <hip/hip_runtime.h>

typedef __attribute__((ext_vector_type(2))) float v2f;
typedef __attribute__((ext_vector_type(8))) float v8f;

#define N_NODES 50000
#define N_EDGES 800000
#define BATCH   4
#define IN_D    64
#define HID_D   64
#define OUT_D   32
#define ROWS    (N_NODES * BATCH)   // 200000 node-batch rows

// ---------------- zero scratch (float4 grid-stride) ----------------
__global__ void gcn_zero_kernel(float4* __restrict__ p, long n4) {
    long i = (long)blockIdx.x * blockDim.x + threadIdx.x;
    long stride = (long)gridDim.x * blockDim.x;
    for (; i < n4; i += stride) p[i] = make_float4(0.f, 0.f, 0.f, 0.f);
}

// ---------------- degree accumulation ----------------
__global__ void gcn_degree_kernel(const int* __restrict__ src, const int* __restrict__ dst,
                                  float* __restrict__ sdeg, float* __restrict__ ddeg) {
    int i = blockIdx.x * blockDim.x + threadIdx.x;
    if (i < N_EDGES) {
        atomicAdd(&sdeg[src[i]], 1.0f);
        atomicAdd(&ddeg[dst[i]], 1.0f);
    }
}

// ---------------- deg -> clip(deg,1)^-0.5 in place ----------------
__global__ void gcn_norm_kernel(float* __restrict__ p, int n) {
    int i = blockIdx.x * blockDim.x + threadIdx.x;
    if (i < n) {
        float v = fmaxf(p[i], 1.0f);
        p[i] = rsqrtf(v);
    }
}

// ---------------- layer-1 edge scatter: agg[dst] += h(src) * s_norm[src] ------
// h is [B, N, 64]; agg is node-major [N, B*64]. 4 edges per 256-thread block,
// 64 lanes per edge, float4 per lane -> b128 loads, f32 L2 atomics.
__global__ void gcn_scatter1_kernel(const float* __restrict__ h,
                                    const int* __restrict__ src, const int* __restrict__ dst,
                                    const float* __restrict__ snorm, float* __restrict__ agg) {
    const int tid = threadIdx.x;
    const int e   = blockIdx.x * 4 + (tid >> 6);
    const int t   = tid & 63;                 // lane within edge
    const int s   = src[e];
    const int d   = dst[e];
    const float sn = snorm[s];
    const int b   = t >> 4;                   // 0..3
    const int f4  = (t & 15) << 2;            // 0,4,...,60
    const float4 v = *(const float4*)(h + ((long)b * N_NODES + s) * IN_D + f4);
    float* ap = agg + (long)d * (BATCH * IN_D) + b * IN_D + f4;
    atomicAdd(ap + 0, v.x * sn);
    atomicAdd(ap + 1, v.y * sn);
    atomicAdd(ap + 2, v.z * sn);
    atomicAdd(ap + 3, v.w * sn);
}

// ---------------- layer-2 edge scatter: contiguous node-major rows ----------
__global__ void gcn_scatter2_kernel(const float* __restrict__ h1,
                                    const int* __restrict__ src, const int* __restrict__ dst,
                                    const float* __restrict__ snorm, float* __restrict__ agg) {
    const int tid = threadIdx.x;
    const int e   = blockIdx.x * 4 + (tid >> 6);
    const int t   = tid & 63;
    const int s   = src[e];
    const int d   = dst[e];
    const float sn = snorm[s];
    const float4 v = *(const float4*)(h1 + (long)s * (BATCH * HID_D) + t * 4);
    float* ap = agg + (long)d * (BATCH * HID_D) + t * 4;
    atomicAdd(ap + 0, v.x * sn);
    atomicAdd(ap + 1, v.y * sn);
    atomicAdd(ap + 2, v.z * sn);
    atomicAdd(ap + 3, v.w * sn);
}

// ---------------- WMMA GEMM: out = relu?( (d_norm * A) @ W + bias ) ----------
// A: [ROWS, 64] node-major aggregate. One wave per 16x16 output tile,
// K=64 via 16x V_WMMA_F32_16X16X4_F32. d_norm folded into A fragment,
// bias (+ReLU) folded into D store. OUT_TRANSPOSE stores [B, N, NCOLS].
template <int NCOLS, bool RELU, bool OUT_TRANSPOSE>
__global__ void gcn_gemm_kernel(const float* __restrict__ A,
                                const float* __restrict__ W,     // [64, NCOLS] row-major
                                const float* __restrict__ bias,  // [NCOLS]
                                const float* __restrict__ dnorm, // [N_NODES]
                                float* __restrict__ out) {
    const int lane   = threadIdx.x & 31;
    const int waveId = blockIdx.x * (blockDim.x >> 5) + (threadIdx.x >> 5);
    const int colTiles = NCOLS / 16;
    const int rowTile  = waveId / colTiles;
    const int colTile  = waveId % colTiles;
    if (rowTile >= ROWS / 16) return;

    const int row0 = rowTile * 16;
    const int col0 = colTile * 16;
    const int m    = lane & 15;   // A row within tile; also D column lane
    const int kb   = lane >> 4;   // K half-select (0 or 1)

    const int row = row0 + m;
    const float dn = dnorm[row >> 2];            // row / BATCH
    const float* arow = A + (long)row * 64;

    v8f c = {};
#pragma unroll
    for (int k0 = 0; k0 < 64; k0 += 4) {
        // A 16x4 f32: VGPR0 = K even, VGPR1 = K odd; half-waves cover K pairs.
        v2f a = *(const v2f*)(arow + k0 + 2 * kb);
        a *= dn;
        // B 4x16 f32: lane = N column, half-wave selects K pair (mirrors A).
        v2f b;
        b.x = W[(k0 + 2 * kb + 0) * NCOLS + col0 + m];
        b.y = W[(k0 + 2 * kb + 1) * NCOLS + col0 + m];
        c = __builtin_amdgcn_wmma_f32_16x16x4_f32(
            /*neg_a=*/false, a, /*neg_b=*/false, b,
            /*c_mod=*/(short)0, c, /*reuse_a=*/false, /*reuse_b=*/false);
    }

    // D 16x16 f32: VGPR i holds M = i (lanes 0-15) / M = i+8 (lanes 16-31), N = lane&15.
    const int col = col0 + m;
    const float bv = bias[col];
#pragma unroll
    for (int i = 0; i < 8; ++i) {
        const int r = row0 + kb * 8 + i;
        float v = c[i] + bv;
        if (RELU) v = fmaxf(v, 0.0f);
        if (OUT_TRANSPOSE) {
            const int n = r >> 2, bb = r & 3;   // row = n*BATCH + b
            out[((long)bb * N_NODES + n) * NCOLS + col] = v;
        } else {
            out[(long)r * NCOLS + col] = v;
        }
    }
}

extern "C" void kernel_launch(void* const* d_in, const int* in_sizes, int n_in,
                              void* d_out, int out_size, void* d_ws, size_t ws_size,
                              hipStream_t stream) {
    (void)in_sizes; (void)n_in; (void)out_size; (void)ws_size;
    const float* h   = (const float*)d_in[0];  // [B, N, 64]
    const float* W1  = (const float*)d_in[1];  // [64, 64]
    const float* b1  = (const float*)d_in[2];  // [64]
    const float* W2  = (const float*)d_in[3];  // [64, 32]
    const float* b2  = (const float*)d_in[4];  // [32]
    const int*   src = (const int*)d_in[5];    // [E]
    const int*   dst = (const int*)d_in[6];    // [E]
    float* out = (float*)d_out;                // [B, N, 32]

    // Workspace layout (floats): snorm[N] | dnorm[N] | agg[ROWS*64] | h1[ROWS*64]
    float* ws    = (float*)d_ws;
    float* snorm = ws;
    float* dnorm = ws + N_NODES;
    float* agg   = ws + 2 * N_NODES;
    float* h1    = agg + (long)ROWS * 64;

    const long aggElems = (long)ROWS * 64;                 // 12.8M floats
    const long zero1_n4 = (2L * N_NODES + aggElems) / 4;   // norms + agg

    // 1) zero norms + agg
    gcn_zero_kernel<<<2048, 256, 0, stream>>>((float4*)ws, zero1_n4);
    // 2) degrees -> 3) deg^-1/2
    gcn_degree_kernel<<<(N_EDGES + 255) / 256, 256, 0, stream>>>(src, dst, snorm, dnorm);
    gcn_norm_kernel<<<(2 * N_NODES + 255) / 256, 256, 0, stream>>>(ws, 2 * N_NODES);
    // 4) layer-1 aggregation (s_norm folded in)
    gcn_scatter1_kernel<<<N_EDGES / 4, 256, 0, stream>>>(h, src, dst, snorm, agg);
    // 5) layer-1 GEMM + bias + ReLU (d_norm folded in): 12500*4 tiles / 8 waves
    gcn_gemm_kernel<HID_D, true, false><<<6250, 256, 0, stream>>>(agg, W1, b1, dnorm, h1);
    // 6) re-zero agg
    gcn_zero_kernel<<<2048, 256, 0, stream>>>((float4*)agg, aggElems / 4);
    // 7) layer-2 aggregation
    gcn_scatter2_kernel<<<N_EDGES / 4, 256, 0, stream>>>(h1, src, dst, snorm, agg);
    // 8) layer-2 GEMM + bias, transposed store to [B, N, 32]: 12500*2 tiles / 8 waves
    gcn_gemm_kernel<OUT_D, false, true><<<3125, 256, 0, stream>>>(agg, W2, b2, dnorm, out);
}